// AutoencODE_stack_63471026700709
// MI455X (gfx1250) — compile-verified
//
#include <hip/hip_runtime.h>

// Kuramoto step, B=64, N=1024, fp32 throughout.
// Bandwidth-bound: couplings (268 MB) streamed exactly once -> ~11.5 us floor @ 23.3 TB/s.
// Dot products on the CDNA5 matrix pipe via V_WMMA_F32_16X16X4_F32 (exact f32).
// D[:,0] accumulates C@sin, D[:,1] accumulates C@cos; degree count rides in VALU.

typedef __attribute__((ext_vector_type(2))) float v2f;
typedef __attribute__((ext_vector_type(8))) float v8f;

#define BATCH 64
#define NDIM  1024
#define WAVES 8                       // waves per block
#define ROWS_PER_BLOCK (WAVES * 16)   // 128

__global__ __launch_bounds__(256) void kuramoto_wmma_kernel(
    const float* __restrict__ C,      // [B, N, N]
    const float* __restrict__ phase,  // [B*N]
    const float* __restrict__ omega,  // [B*N]
    float* __restrict__ out)          // [B*N]
{
    // one pool: [0..N) = sin, [N..2N) = cos, [2N..2N+8) = zero pad for inactive B lanes
    __shared__ __align__(16) float bpool[2 * NDIM + 8];
    __shared__ __align__(16) float red[WAVES * 32];      // cs/cc staging per wave
    __shared__ __align__(16) float cnt_lds[WAVES * 32];  // per-lane nonzero counts

    const int tid  = threadIdx.x;
    const int wave = tid >> 5;
    const int lane = tid & 31;

    const int b       = blockIdx.x >> 3;                    // batch
    const int slab    = blockIdx.x & 7;                     // 128-row slab
    const int rowBase = slab * ROWS_PER_BLOCK + wave * 16;  // wave's 16 rows (in-batch)

    // ---- sin/cos of this batch's 1024 phases -> LDS (coalesced, computed in-block) ----
    {
        const float4 p4 = ((const float4*)(phase + b * NDIM))[tid];  // 256*4 = 1024
        float4 s4, c4;
        s4.x = sinf(p4.x); s4.y = sinf(p4.y); s4.z = sinf(p4.z); s4.w = sinf(p4.w);
        c4.x = cosf(p4.x); c4.y = cosf(p4.y); c4.z = cosf(p4.z); c4.w = cosf(p4.w);
        ((float4*)bpool)[tid]          = s4;
        ((float4*)(bpool + NDIM))[tid] = c4;
    }
    if (tid < 8) bpool[2 * NDIM + tid] = 0.0f;
    __syncthreads();

    // ---- A-fragment addressing: chunk-pair scheme, one b128 per lane per 8 columns ----
    // lanes 0-15 (row m) own cols col..col+3; lanes 16-31 (row m) own cols col+4..col+7.
    // WMMA even chunk consumes (.x,.y), odd chunk (.z,.w); B rows are permuted to match.
    const int mrow = lane & 15;
    const int hi   = lane >> 4;
    const float* rowPtr =
        C + (size_t)(b * NDIM + rowBase + mrow) * NDIM + 4 * hi;

    // ---- B-fragment per-lane pointer: N=0 col -> sin, N=1 col -> cos, rest -> zero pad ----
    const bool active = (mrow < 2);
    int bidx = (mrow == 0 ? 0 : (mrow == 1 ? NDIM : 2 * NDIM)) + (active ? 4 * hi : 0);
    const int step2 = active ? 2 : 0;   // even->odd chunk advance (0 keeps zero lanes on pad)
    const int step8 = active ? 8 : 0;   // pair->pair advance

    v8f  acc0 = {};     // even chunks: D[:,0]=cs, D[:,1]=cc partials
    v8f  acc1 = {};     // odd  chunks
    float cnt = 0.0f;   // per-lane nonzero count (4 elements / pair)

    #pragma unroll 4
    for (int col = 0; col < NDIM; col += 8) {
        const float4 a = *(const float4*)(rowPtr + col);   // global b128, used once

        v2f a01; a01.x = a.x; a01.y = a.y;
        v2f a23; a23.x = a.z; a23.y = a.w;

        const v2f b0 = *(const v2f*)(bpool + bidx);          // ds b64 (broadcast/pad)
        const v2f b1 = *(const v2f*)(bpool + bidx + step2);  // ds b64
        bidx += step8;

        acc0 = __builtin_amdgcn_wmma_f32_16x16x4_f32(
            false, a01, false, b0, (short)0, acc0, false, false);
        acc1 = __builtin_amdgcn_wmma_f32_16x16x4_f32(
            false, a23, false, b1, (short)0, acc1, false, false);

        cnt += (a.x != 0.0f) ? 1.0f : 0.0f;
        cnt += (a.y != 0.0f) ? 1.0f : 0.0f;
        cnt += (a.z != 0.0f) ? 1.0f : 0.0f;
        cnt += (a.w != 0.0f) ? 1.0f : 0.0f;
    }

    const v8f acc = acc0 + acc1;

    // ---- extract D columns 0 (cs) and 1 (cc) ----
    // C/D layout: acc[r] at lane L is element (M = r + 8*(L>=16), N = L&15)
    if (mrow < 2) {   // lanes 0,1,16,17
        const int mh = hi * 8;
        #pragma unroll
        for (int r = 0; r < 8; ++r)
            red[wave * 32 + (mh + r) * 2 + mrow] = acc[r];
    }
    cnt_lds[wave * 32 + lane] = cnt;
    __syncthreads();

    // ---- final combine: (cs*cos_i - cc*sin_i)/n + omega ----
    if (lane < 16) {
        const int m          = lane;
        const int rowInBatch = rowBase + m;
        const float cs = red[wave * 32 + m * 2 + 0];
        const float cc = red[wave * 32 + m * 2 + 1];
        const float n  = cnt_lds[wave * 32 + m] + cnt_lds[wave * 32 + m + 16];
        const float si = bpool[rowInBatch];
        const float ci = bpool[NDIM + rowInBatch];
        const int   gi = b * NDIM + rowInBatch;
        out[gi] = (cs * ci - cc * si) / n + omega[gi];
    }
}

extern "C" void kernel_launch(void* const* d_in, const int* in_sizes, int n_in,
                              void* d_out, int out_size, void* d_ws, size_t ws_size,
                              hipStream_t stream) {
    // inputs: d_in[0] = t (unused), d_in[1] = phase, d_in[2] = couplings, d_in[3] = omega
    const float* phase = (const float*)d_in[1];
    const float* coup  = (const float*)d_in[2];
    const float* omega = (const float*)d_in[3];
    float* out = (float*)d_out;

    dim3 grid(BATCH * (NDIM / ROWS_PER_BLOCK));  // 512 blocks
    dim3 block(256);                             // 8 waves (wave32)
    kuramoto_wmma_kernel<<<grid, block, 0, stream>>>(coup, phase, omega, out);
}